// RawAudioProsodyExtractor_67645734912347
// MI455X (gfx1250) — compile-verified
//
#include <hip/hip_runtime.h>
#include <hip/hip_bf16.h>

// ---------------------------------------------------------------------------
// RawAudioProsodyExtractor for MI455X (gfx1250, wave32).
//   k_frame_stats : per-frame RMS/ZCR/amp + autocorr-peak F0 (direct autocorr,
//                   identical to the FFT-based linear autocorr in the ref).
//   k_hnr_ac      : full-signal autocorr lags 50..199 (direct dot products).
//   k_finalize    : per-batch reductions, jitter (parallel chunk scan +
//                   monoid combine), shimmer, HNR percentile, metrics+feats.
//   k_project     : MLP projection using V_WMMA_F32_16X16X4_F32 (fp32 WMMA).
// ---------------------------------------------------------------------------

#define NSAMP 160000
#define NF    998
#define FRAME 400
#define HOP   160
#define SBUFW 720   // FRAME + 320 zero pad -> no bounds check in autocorr

typedef float v2f __attribute__((ext_vector_type(2)));
typedef float v8f __attribute__((ext_vector_type(8)));

__device__ __forceinline__ float waveSum(float v) {
    for (int o = 16; o; o >>= 1) v += __shfl_xor(v, o, 32);
    return v;
}
__device__ __forceinline__ float waveMax(float v) {
    for (int o = 16; o; o >>= 1) v = fmaxf(v, __shfl_xor(v, o, 32));
    return v;
}

__device__ __forceinline__ float blkSum(float v, float* red) {
    int t = threadIdx.x;
    red[t] = v; __syncthreads();
    for (int s = 128; s > 0; s >>= 1) { if (t < s) red[t] += red[t + s]; __syncthreads(); }
    float r = red[0]; __syncthreads();
    return r;
}
__device__ __forceinline__ float blkMax(float v, float* red) {
    int t = threadIdx.x;
    red[t] = v; __syncthreads();
    for (int s = 128; s > 0; s >>= 1) { if (t < s) red[t] = fmaxf(red[t], red[t + s]); __syncthreads(); }
    float r = red[0]; __syncthreads();
    return r;
}

// ---------------------------------------------------------------------------
// Kernel 1: one wave (32 lanes) per frame.  8 waves / block of 256 threads.
// grid = (ceil(NF/8), B)
// ---------------------------------------------------------------------------
__global__ void k_frame_stats(const float* __restrict__ audio,
                              float* __restrict__ rms, float* __restrict__ zcr,
                              float* __restrict__ amp, float* __restrict__ f0a,
                              float* __restrict__ valid) {
    const int wid  = threadIdx.x >> 5;
    const int lane = threadIdx.x & 31;
    const int b    = blockIdx.y;
    const int f    = blockIdx.x * 8 + wid;
    const bool active = (f < NF);
    const int fc   = active ? f : 0;

    __shared__ float sb[8 * SBUFW];
    float* s = sb + wid * SBUFW;
    const float* src = audio + (size_t)b * NSAMP + fc * HOP;
    for (int j = lane; j < FRAME; j += 32) s[j] = src[j];
    for (int j = FRAME + lane; j < SBUFW; j += 32) s[j] = 0.f;
    __syncthreads();

    // --- sum of squares, ZCR numerator, peak amplitude ---
    float sumsq = 0.f, zsum = 0.f, mx = 0.f;
    for (int j = lane; j < FRAME; j += 32) {
        float x = s[j];
        sumsq += x * x;
        mx = fmaxf(mx, fabsf(x));
        if (j < FRAME - 1) {
            float x1 = s[j + 1];
            float sg0 = (x  > 0.f) ? 1.f : ((x  < 0.f) ? -1.f : 0.f);
            float sg1 = (x1 > 0.f) ? 1.f : ((x1 < 0.f) ? -1.f : 0.f);
            zsum += fabsf(sg1 - sg0);
        }
    }
    sumsq = waveSum(sumsq);
    zsum  = waveSum(zsum);
    mx    = waveMax(mx);

    // --- direct linear autocorrelation, lags 32..319 (9 lags per lane) ---
    float acc[9];
#pragma unroll
    for (int t = 0; t < 9; t++) acc[t] = 0.f;
    for (int i = 0; i < FRAME; i++) {
        float xi = s[i];
#pragma unroll
        for (int t = 0; t < 9; t++)
            acc[t] += xi * s[i + 32 + lane + 32 * t];   // zero pad covers tail
    }
    // first-max (lowest lag) selection, matching np.argmax tie-breaking
    float bestv = acc[0]; int bestl = 32 + lane;
#pragma unroll
    for (int t = 1; t < 9; t++) {
        int l = 32 + lane + 32 * t;
        if (acc[t] > bestv) { bestv = acc[t]; bestl = l; }
    }
    for (int o = 16; o; o >>= 1) {
        float ov = __shfl_xor(bestv, o, 32);
        int   ol = __shfl_xor(bestl, o, 32);
        if (ov > bestv || (ov == bestv && ol < bestl)) { bestv = ov; bestl = ol; }
    }

    if (active && lane == 0) {
        int idx = b * NF + f;
        rms[idx] = sqrtf(sumsq / (float)FRAME);
        zcr[idx] = zsum / (2.f * FRAME);
        amp[idx] = mx;
        float f0 = 16000.f / (float)bestl;
        f0a[idx] = f0;
        float ac0 = sumsq;                      // ac[0] == sum x^2
        valid[idx] = ((bestv > 0.3f * ac0) && (f0 > 50.f) && (f0 < 500.f)) ? 1.f : 0.f;
    }
}

// ---------------------------------------------------------------------------
// Kernel 2: HNR autocorrelation, one block per (lag, batch). grid = (150, B)
// ---------------------------------------------------------------------------
__global__ void k_hnr_ac(const float* __restrict__ audio, float* __restrict__ hac) {
    const int lag = 50 + blockIdx.x;
    const int b   = blockIdx.y;
    const float* a = audio + (size_t)b * NSAMP;
    float sum = 0.f;
    for (int i = threadIdx.x; i < NSAMP - lag; i += 256)
        sum += a[i] * a[i + lag];
    __shared__ float red[256];
    int t = threadIdx.x;
    red[t] = sum; __syncthreads();
    for (int s = 128; s > 0; s >>= 1) { if (t < s) red[t] += red[t + s]; __syncthreads(); }
    if (t == 0) hac[b * 150 + blockIdx.x] = red[0];
}

// ---------------------------------------------------------------------------
// Kernel 3: per-batch finalize.  grid = B, block = 256.
// ---------------------------------------------------------------------------
__global__ void k_finalize(const float* __restrict__ audio,
                           const float* __restrict__ rms, const float* __restrict__ zcr,
                           const float* __restrict__ amp, const float* __restrict__ f0a,
                           const float* __restrict__ validm, const float* __restrict__ hac,
                           float* __restrict__ feats, float* __restrict__ metrics_out) {
    const int b = blockIdx.x;
    const int t = threadIdx.x;
    const float* a = audio + (size_t)b * NSAMP;

    __shared__ float red[256];
    __shared__ int   ccnt[256];
    __shared__ float cpart[256], cf0[256], cf1[256], cl0[256], cl1[256];
    __shared__ float acb[150];

    // --- threshold = 0.3 * max|a| ---
    float mx = 0.f;
    for (int i = t; i < NSAMP; i += 256) mx = fmaxf(mx, fabsf(a[i]));
    mx = blkMax(mx, red);
    const float thr = 0.3f * mx;

    // --- jitter: chunked rising-crossing scan (summary per 625-sample chunk) ---
    {
        int iStart = 1 + t * 625;
        int iEnd   = iStart + 625; if (iEnd > NSAMP) iEnd = NSAMP;
        int cnt = 0; float part = 0.f, f0v = 0.f, f1v = 0.f, l0v = 0.f, l1v = 0.f;
        if (iStart < NSAMP) {
            float prev = a[iStart - 1];
            for (int i = iStart; i < iEnd; i++) {
                float cur = a[i];
                if (prev < thr && cur >= thr) {
                    float p = (float)i;
                    if (cnt >= 2) part += fabsf(p - 2.f * l1v + l0v);
                    if (cnt == 0) f0v = p; else if (cnt == 1) f1v = p;
                    l0v = l1v; l1v = p;
                    cnt++;
                }
                prev = cur;
            }
        }
        ccnt[t] = cnt; cpart[t] = part; cf0[t] = f0v; cf1[t] = f1v; cl0[t] = l0v; cl1[t] = l1v;
    }
    __syncthreads();

    // --- frame-level reductions ---
    float sR = 0, sR2 = 0, sZ = 0, sZ2 = 0, sF = 0, sF2 = 0, sC = 0, sV = 0;
    for (int i = t; i < NF; i += 256) {
        int idx = b * NF + i;
        float r = rms[idx], z = zcr[idx], f = f0a[idx], m = validm[idx];
        sR += r; sR2 += r * r; sZ += z; sZ2 += z * z;
        sF += f * m; sF2 += f * f * m; sC += m;
        sV += ((r > 0.01f) && (z < 0.3f)) ? 1.f : 0.f;
    }
    sR = blkSum(sR, red); sR2 = blkSum(sR2, red);
    sZ = blkSum(sZ, red); sZ2 = blkSum(sZ2, red);
    sF = blkSum(sF, red); sF2 = blkSum(sF2, red);
    sC = blkSum(sC, red); sV  = blkSum(sV, red);

    if (t < 150) acb[t] = hac[b * 150 + t];
    __syncthreads();

    if (t == 0) {
        float e_mean = sR / NF;
        float e_std  = sqrtf(fmaxf(sR2 / NF - e_mean * e_mean, 0.f));
        float z_mean = sZ / NF;
        float z_std  = sqrtf(fmaxf(sZ2 / NF - z_mean * z_mean, 0.f));
        float c      = fmaxf(sC, 1.f);
        float f0_mean = sF / c;
        float f0_std  = sqrtf(fmaxf(sF2 / c - f0_mean * f0_mean, 0.f));
        if (sC <= 0.f) { f0_mean = 0.f; f0_std = 0.f; }
        float voicing = sV / NF;

        // --- jitter: combine the 256 chunk summaries in order ---
        long cnt = 0; float part = 0.f, f0v = 0.f, f1v = 0.f, l0v = 0.f, l1v = 0.f;
        for (int q = 0; q < 256; q++) {
            int rc = ccnt[q]; if (rc == 0) continue;
            float rp = cpart[q], rf0 = cf0[q], rf1 = cf1[q], rl0 = cl0[q], rl1 = cl1[q];
            if (cnt >= 2 && rc >= 1) part += fabsf(rf0 - 2.f * l1v + l0v);
            if (cnt >= 1 && rc >= 2) part += fabsf(rf1 - 2.f * rf0 + l1v);
            part += rp;
            if (cnt == 0)      { f0v = rf0; if (rc >= 2) f1v = rf1; }
            else if (cnt == 1) { f1v = rf0; }
            if (rc >= 2) { l0v = rl0; l1v = rl1; }
            else         { l0v = l1v; l1v = rf0; }
            cnt += rc;
        }
        float jitter = 0.f;
        {
            long nper = cnt - 1;
            float mean_period = (nper >= 1) ? (l1v - f0v) / (float)nper : 0.f;
            float mean_pdiff  = (nper >= 2) ? part / (float)(nper - 1) : 0.f;
            if (nper >= 2 && mean_period > 0.f)
                jitter = fminf(fmaxf(mean_pdiff / fmaxf(mean_period, 1e-12f), 0.f), 1.f);
            if (f0_mean < 50.f || f0_mean > 500.f) jitter = 0.f;
        }

        // --- shimmer: serial over 998 frame amps (cheap) ---
        int ka = 0; float sumAmp = 0.f, sumAd = 0.f, prevA = 0.f;
        for (int i = 0; i < NF; i++) {
            float av = amp[b * NF + i];
            if (av > 0.01f) {
                if (ka > 0) sumAd += fabsf(av - prevA);
                sumAmp += av; prevA = av; ka++;
            }
        }
        float mean_amp   = sumAmp / fmaxf((float)ka, 1.f);
        float mean_adiff = sumAd  / fmaxf((float)(ka - 1), 1.f);
        float shimmer = 0.f;
        if (ka >= 2 && mean_amp > 0.f)
            shimmer = fminf(fmaxf(mean_adiff / fmaxf(mean_amp, 1e-12f), 0.f), 1.f);

        // --- HNR: peak + 10th percentile (linear interp at idx 14.9) ---
        float sig_peak = -1e30f;
        for (int i = 0; i < 150; i++) sig_peak = fmaxf(sig_peak, acb[i]);
        for (int j = 0; j < 16; j++) {        // partial selection sort (ascending)
            int mi = j;
            for (int i = j + 1; i < 150; i++) if (acb[i] < acb[mi]) mi = i;
            float tmp = acb[j]; acb[j] = acb[mi]; acb[mi] = tmp;
        }
        float noise_floor = acb[14] + 0.9f * (acb[15] - acb[14]);
        float hnr = 0.f;
        if (noise_floor > 0.f)
            hnr = fminf(fmaxf(sig_peak / fmaxf(noise_floor, 1e-30f) / 100.f, 0.f), 1.f);

        float mets[10] = { f0_mean, f0_std, e_mean, e_std, z_mean, z_std,
                           jitter, shimmer, hnr, voicing };
        float fts[10]  = { f0_mean / 500.f, f0_std / 100.f, e_mean, e_std, z_mean, z_std,
                           jitter, shimmer, hnr, voicing };
        for (int i = 0; i < 10; i++) {
            metrics_out[b * 10 + i] = mets[i];
            feats[b * 10 + i]       = fts[i];
        }
    }
}

// ---------------------------------------------------------------------------
// Kernel 4: MLP projection with fp32 WMMA (V_WMMA_F32_16X16X4_F32).
// Block = 1024 threads = 32 waves; one 16x16 tile per wave.
//   GEMM1: [128x10] @ W1^T[10x64]  (K padded to 12, 3 k-steps of 4)
//   LN+ReLU per row in LDS
//   GEMM2: [128x64] @ W2^T[64x128] (16 k-steps, 2 tiles per wave)
// Fragment layout per ISA 7.12.2:
//   A 16x4 f32: 2 VGPRs, lane half (>=16) selects K {0,1} vs {2,3}, M = lane&15
//   B 4x16 f32: mirrored (K from half, N = lane&15)
//   C/D 16x16:  8 VGPRs, row = r + 8*half, col = lane&15
// ---------------------------------------------------------------------------
__global__ void __launch_bounds__(1024)
k_project(const float* __restrict__ feats,
          const float* __restrict__ W1, const float* __restrict__ b1,
          const float* __restrict__ gma, const float* __restrict__ bta,
          const float* __restrict__ W2, const float* __restrict__ b2,
          float* __restrict__ out) {
    __shared__ float H[128 * 64];
    const int w    = threadIdx.x >> 5;
    const int lane = threadIdx.x & 31;
    const int half = lane >> 4;
    const int l16  = lane & 15;

    // ---------------- GEMM1 + bias ----------------
    {
        const int mt = w >> 2, nt = w & 3;     // 8 x 4 tiles
        v8f acc = {0.f, 0.f, 0.f, 0.f, 0.f, 0.f, 0.f, 0.f};
#pragma unroll
        for (int ks = 0; ks < 3; ks++) {
            const int k0 = ks * 4 + half * 2;
            v2f av, bv;
            av.x = (k0     < 10) ? feats[(mt * 16 + l16) * 10 + k0]     : 0.f;
            av.y = (k0 + 1 < 10) ? feats[(mt * 16 + l16) * 10 + k0 + 1] : 0.f;
            bv.x = (k0     < 10) ? W1[(nt * 16 + l16) * 10 + k0]        : 0.f;
            bv.y = (k0 + 1 < 10) ? W1[(nt * 16 + l16) * 10 + k0 + 1]    : 0.f;
            acc = __builtin_amdgcn_wmma_f32_16x16x4_f32(
                      false, av, false, bv, (short)0, acc, false, false);
        }
        const int col = nt * 16 + l16;
        const float bb = b1[col];
#pragma unroll
        for (int r = 0; r < 8; r++) {
            const int row = mt * 16 + r + 8 * half;
            H[row * 64 + col] = acc[r] + bb;
        }
    }
    __syncthreads();

    // ---------------- LayerNorm + ReLU (one row per thread) ----------------
    if (threadIdx.x < 128) {
        const int row = threadIdx.x;
        float s = 0.f, s2 = 0.f;
        for (int c = 0; c < 64; c++) { float v = H[row * 64 + c]; s += v; s2 += v * v; }
        const float mu  = s / 64.f;
        const float var = fmaxf(s2 / 64.f - mu * mu, 0.f);
        const float inv = rsqrtf(var + 1e-5f);
        for (int c = 0; c < 64; c++) {
            float v = (H[row * 64 + c] - mu) * inv * gma[c] + bta[c];
            H[row * 64 + c] = fmaxf(v, 0.f);
        }
    }
    __syncthreads();

    // ---------------- GEMM2 + bias ----------------
#pragma unroll
    for (int tt = 0; tt < 2; tt++) {
        const int tile = w + 32 * tt;          // 64 tiles: 8 x 8
        const int mt = tile >> 3, nt = tile & 7;
        v8f acc = {0.f, 0.f, 0.f, 0.f, 0.f, 0.f, 0.f, 0.f};
#pragma unroll
        for (int ks = 0; ks < 16; ks++) {
            const int k0 = ks * 4 + half * 2;
            v2f av, bv;
            av.x = H[(mt * 16 + l16) * 64 + k0];
            av.y = H[(mt * 16 + l16) * 64 + k0 + 1];
            bv.x = W2[(nt * 16 + l16) * 64 + k0];
            bv.y = W2[(nt * 16 + l16) * 64 + k0 + 1];
            acc = __builtin_amdgcn_wmma_f32_16x16x4_f32(
                      false, av, false, bv, (short)0, acc, false, false);
        }
        const int col = nt * 16 + l16;
        const float bb = b2[col];
#pragma unroll
        for (int r = 0; r < 8; r++) {
            const int row = mt * 16 + r + 8 * half;
            out[row * 128 + col] = acc[r] + bb;
        }
    }
}

// ---------------------------------------------------------------------------
extern "C" void kernel_launch(void* const* d_in, const int* in_sizes, int n_in,
                              void* d_out, int out_size, void* d_ws, size_t ws_size,
                              hipStream_t stream) {
    (void)in_sizes; (void)n_in; (void)out_size; (void)ws_size;
    const float* audio = (const float*)d_in[0];
    const float* W1    = (const float*)d_in[1];
    const float* b1    = (const float*)d_in[2];
    const float* gma   = (const float*)d_in[3];
    const float* bta   = (const float*)d_in[4];
    const float* W2    = (const float*)d_in[5];
    const float* b2    = (const float*)d_in[6];
    float* out = (float*)d_out;
    float* ws  = (float*)d_ws;

    const int B = 128;
    // workspace layout (floats): total 659,200 (~2.6 MB)
    float* feats = ws;                    // B*10
    float* rms   = feats + B * 10;        // B*NF
    float* zcr   = rms   + B * NF;
    float* amp   = zcr   + B * NF;
    float* f0a   = amp   + B * NF;
    float* valid = f0a   + B * NF;
    float* hac   = valid + B * NF;        // B*150

    dim3 g1((NF + 7) / 8, B);
    k_frame_stats<<<g1, 256, 0, stream>>>(audio, rms, zcr, amp, f0a, valid);

    dim3 g2(150, B);
    k_hnr_ac<<<g2, 256, 0, stream>>>(audio, hac);

    k_finalize<<<B, 256, 0, stream>>>(audio, rms, zcr, amp, f0a, valid, hac,
                                      feats, out + 128 * 128);

    k_project<<<1, 1024, 0, stream>>>(feats, W1, b1, gma, bta, W2, b2, out);
}